// GaussianSplatting_87531433492843
// MI455X (gfx1250) — compile-verified
//
#include <hip/hip_runtime.h>

#define IMG_H 512
#define IMG_W 512
#define NGAUSS 10000
#define N_VIEWS 4
#define PATCH 40

typedef __attribute__((ext_vector_type(2))) float v2f;
typedef __attribute__((ext_vector_type(8))) float v8f;

__global__ void gs_zero_kernel(float4* __restrict__ p, int n4) {
    int i = blockIdx.x * blockDim.x + threadIdx.x;
    if (i < n4) p[i] = make_float4(0.f, 0.f, 0.f, 0.f);
}

// One wave32 per (view, gaussian). 40x40 patch covered by 3x3 tiles of 16x16,
// each tile's exponent matrix computed with one v_wmma_f32_16x16x4_f32.
__global__ __launch_bounds__(32) void gs_splat_kernel(
    const float* __restrict__ poses,      // (4,4,4)
    const float* __restrict__ K,          // (3,3)
    const float* __restrict__ means,      // (N,3)
    const float* __restrict__ log_scales, // (N,3)
    const float* __restrict__ quats,      // (N,4)
    const float* __restrict__ shs,        // (N,3,16)
    const float* __restrict__ opac,       // (N,1)
    float* __restrict__ out)              // (4,H,W,3)
{
    const int gv   = blockIdx.x;
    const int n    = gv % NGAUSS;
    const int view = gv / NGAUSS;
    const int lane = threadIdx.x;

    // ---- projection (uniform across wave; compiler scalarizes) ----
    const float* P = poses + view * 16;
    float mx = means[n*3+0], my = means[n*3+1], mz = means[n*3+2];
    float Xc = P[0]*mx + P[1]*my + P[2]*mz  + P[3];
    float Yc = P[4]*mx + P[5]*my + P[6]*mz  + P[7];
    float Zc = P[8]*mx + P[9]*my + P[10]*mz + P[11];
    float px = K[0]*Xc + K[1]*Yc + K[2]*Zc;
    float py = K[3]*Xc + K[4]*Yc + K[5]*Zc;
    float pz = K[6]*Xc + K[7]*Yc + K[8]*Zc;
    float invz = 1.0f / (pz + 1e-8f);
    float uvx = px * invz, uvy = py * invz;
    float uf = truncf(uvx), vf = truncf(uvy);
    bool valid = (Zc >= 0.1f) && (uf >= 0.f) && (uf < (float)IMG_W)
                             && (vf >= 0.f) && (vf < (float)IMG_H);
    if (!valid) return;   // wave-uniform: EXEC stays all-1s on the WMMA path
    int u0 = (int)uf, v0 = (int)vf;

    // ---- 2D covariance inverse coefficients ----
    float sx = __expf(log_scales[n*3+0]);
    float sy = __expf(log_scales[n*3+1]);
    float sz = __expf(log_scales[n*3+2]);
    float qw = quats[n*4+0], qx = quats[n*4+1], qy = quats[n*4+2], qz = quats[n*4+3];
    float r00 = 1.f - 2.f*(qy*qy + qz*qz);
    float r01 = 2.f*(qx*qy - qw*qz);
    float r02 = 2.f*(qx*qz + qw*qy);
    float r10 = 2.f*(qx*qy + qw*qz);
    float r11 = 1.f - 2.f*(qx*qx + qz*qz);
    float r12 = 2.f*(qy*qz - qw*qx);
    float c00 = sx*r00*r00 + sy*r01*r01 + sz*r02*r02;
    float c01 = sx*r00*r10 + sy*r01*r11 + sz*r02*r12;
    float c11 = sx*r10*r10 + sy*r11*r11 + sz*r12*r12;
    float det  = c00*c11 - c01*c01;
    float idet = 1.0f / det;
    float cA = -0.5f * c11 * idet;     // coeff of dx^2
    float cD = -0.5f * c00 * idet;     // coeff of dy^2
    float cB =  c01 * idet;            // coeff of dx*dy (= -0.5*(invB+invC))

    // ---- premultiplied color = sigmoid(op) * sigmoid(sh0) ----
    float opv = 1.f / (1.f + __expf(-opac[n]));
    float col0 = opv / (1.f + __expf(-shs[n*48 +  0]));
    float col1 = opv / (1.f + __expf(-shs[n*48 + 16]));
    float col2 = opv / (1.f + __expf(-shs[n*48 + 32]));

    const int half = lane >> 4;   // 0: K0/K1 slots, 1: K2/K3 slots
    const int idx  = lane & 15;

    for (int ty = 0; ty < 3; ++ty) {
        // A fragment (16x4 f32): A[m] = [1, cD*dy^2, dy, 0], dy for row m = idx
        float dy = (float)(v0 - PATCH/2 + ty*16 + idx) - uvy;
        v2f a;
        if (half == 0) { a.x = 1.0f;  a.y = cD * dy * dy; }   // K=0, K=1
        else           { a.x = dy;    a.y = 0.0f;         }   // K=2, K=3

        for (int tx = 0; tx < 3; ++tx) {
            // B fragment (4x16 f32): B[:,n] = [cA*dx^2, 1, cB*dx, 0]
            float dx = (float)(u0 - PATCH/2 + tx*16 + idx) - uvx;
            v2f b;
            if (half == 0) { b.x = cA * dx * dx; b.y = 1.0f; }  // K=0, K=1
            else           { b.x = cB * dx;      b.y = 0.0f; }  // K=2, K=3

            v8f acc = {};
            acc = __builtin_amdgcn_wmma_f32_16x16x4_f32(
                /*neg_a=*/false, a, /*neg_b=*/false, b,
                /*c_mod=*/(short)0, acc, /*reuse_a=*/false, /*reuse_b=*/false);

            // C/D layout: lane<16 -> (M=r, N=lane); lane>=16 -> (M=r+8, N=lane-16)
            #pragma unroll
            for (int r = 0; r < 8; ++r) {
                int rowm = r + half * 8;
                int pr = ty * 16 + rowm;
                int pc = tx * 16 + idx;
                if (pr >= PATCH || pc >= PATCH) continue;
                int y = v0 - PATCH/2 + pr;
                int x = u0 - PATCH/2 + pc;
                float expo = fminf(0.0f, fmaxf(-10.0f, acc[r]));
                float g = __expf(expo);
                if (y >= 0 && y < IMG_H && x >= 0 && x < IMG_W && g > 0.001f) {
                    float* dst = out + (((size_t)view * IMG_H + y) * IMG_W + x) * 3;
                    atomicAdd(dst + 0, g * col0);
                    atomicAdd(dst + 1, g * col1);
                    atomicAdd(dst + 2, g * col2);
                }
            }
        }
    }
}

extern "C" void kernel_launch(void* const* d_in, const int* in_sizes, int n_in,
                              void* d_out, int out_size, void* d_ws, size_t ws_size,
                              hipStream_t stream) {
    const float* poses  = (const float*)d_in[0];
    const float* intr   = (const float*)d_in[1];
    const float* means  = (const float*)d_in[2];
    const float* lscal  = (const float*)d_in[3];
    const float* quats  = (const float*)d_in[4];
    const float* shs    = (const float*)d_in[5];
    const float* opac   = (const float*)d_in[6];
    float* out = (float*)d_out;

    // out_size = 4*512*512*3 = 3,145,728 (divisible by 4)
    int n4 = out_size / 4;
    gs_zero_kernel<<<(n4 + 255) / 256, 256, 0, stream>>>((float4*)out, n4);

    gs_splat_kernel<<<N_VIEWS * NGAUSS, 32, 0, stream>>>(
        poses, intr, means, lscal, quats, shs, opac, out);
}